// NoiseModelLoss_90795608638367
// MI455X (gfx1250) — compile-verified
//
#include <hip/hip_runtime.h>
#include <math.h>

// ---------------- problem constants (from reference) ----------------
#define T_LEN  262144
#define B_N    32
#define K_N    64
#define W_R    20          // peakiness window radius
#define BDIM   256
#define CHUNK  4096
#define NCHUNK (T_LEN / CHUNK)     // 64
#define EPT    (CHUNK / BDIM)      // 16 outputs per thread
#define SPAN   (CHUNK + 2 * W_R)   // 4136 LDS floats
#define PI_F   3.14159265358979f

typedef unsigned int u32x4 __attribute__((ext_vector_type(4)));
typedef int          i32x8 __attribute__((ext_vector_type(8)));
typedef int          i32x4 __attribute__((ext_vector_type(4)));
typedef float        v2f   __attribute__((ext_vector_type(2)));
typedef float        v8f   __attribute__((ext_vector_type(8)));

// =====================================================================
// Kernel 1: fused [B,T] pass.
//   Per block (one 4096-elem chunk of one row):
//     - TDM async-load heatmap chunk+halo into LDS (tensor_load_to_lds)
//     - mask-multiply in LDS  (mh = heatmap * mask)
//     - 41-wide sliding max via common/suffix/prefix decomposition
//     - CenterNet focal terms from logits/warmstart (overlaps the DMA)
//   Emits 4 partial sums: focal_sum, num_pos, sum(mh), sum(mh*ismax)
// =====================================================================
__global__ __launch_bounds__(BDIM)
void fused_row_pass(const float* __restrict__ hm,
                    const int*   __restrict__ mask,
                    const float* __restrict__ logits,
                    const float* __restrict__ warm,
                    float* __restrict__ partials)
{
  __shared__ float smh[SPAN];
  __shared__ float red[4 * BDIM];

  const int  chunk  = blockIdx.x;
  const int  b      = blockIdx.y;
  const int  tid    = threadIdx.x;
  const int  start  = chunk * CHUNK;
  const long rowoff = (long)b * T_LEN;

  const int s0 = start - W_R;
  const int s1 = start + CHUNK + W_R;
  const int c0 = (s0 < 0) ? 0 : s0;
  const int c1 = (s1 > T_LEN) ? T_LEN : s1;
  const int lz = c0 - s0;   // left  clamp (0 or 20) -> zero-pad
  const int rz = s1 - c1;   // right clamp (0 or 20) -> zero-pad

  // zero-fill out-of-row halo slots (disjoint from TDM target -> no race)
  if (tid < lz) smh[tid] = 0.0f;
  if (tid < rz) smh[SPAN - 1 - tid] = 0.0f;

  // ---- Tensor Data Mover: DMA heatmap span [c0,c1) -> LDS (wave 0 only;
  // TDM ignores EXEC, TENSORcnt is per-wave) ----
  if (tid == 0) {
    unsigned long long ga = (unsigned long long)(const void*)(hm + rowoff + c0);
    unsigned lds = (unsigned)(uintptr_t)(void*)&smh[lz]; // flat addr low 32b = LDS offset
    unsigned n   = (unsigned)(c1 - c0);                  // <= 4136, fits tile_dim0
    u32x4 g0; i32x8 g1; i32x4 g2; i32x4 g3; i32x8 g4;
    // D# group 0: count=1 | lds_addr | global_addr[56:0] | type=2
    g0[0] = 1u;
    g0[1] = lds;
    g0[2] = (unsigned)(ga & 0xFFFFFFFFu);
    g0[3] = (unsigned)((ga >> 32) & 0x01FFFFFFu) | (2u << 30);
    // D# group 1: data_size=4B, tensor_dim0=n, tensor_dim1=1, tile_dim0=n,
    //             tile_dim1=1, tensor_dim0_stride=n
    g1[0] = (int)(2u << 16);                               // data_size code 2
    g1[1] = (int)((n & 0xFFFFu) << 16);                    // tensor_dim0[15:0] @ [63:48]
    g1[2] = (int)(((n >> 16) & 0xFFFFu) | (1u << 16));     // tensor_dim0[31:16] | tensor_dim1 lo = 1
    g1[3] = (int)((n & 0xFFFFu) << 16);                    // tensor_dim1 hi = 0 | tile_dim0 = n
    g1[4] = 1;                                             // tile_dim1 = 1, tile_dim2 = 0
    g1[5] = (int)n;                                        // tensor_dim0_stride lo32
    g1[6] = 0;
    g1[7] = 0;
    g2 = (i32x4){0, 0, 0, 0};
    g3 = (i32x4){0, 0, 0, 0};
    g4 = (i32x8){0, 0, 0, 0, 0, 0, 0, 0};
    __builtin_amdgcn_tensor_load_to_lds(g0, g1, g2, g3, g4, 0);
  }

  // ---- focal inputs for this thread's 16 outputs (overlaps the DMA) ----
  const int tbase = start + tid * EPT;
  float xs[EPT], gs[EPT], ms[EPT];
  {
    const float4* xp = (const float4*)(logits + rowoff + tbase);
    const float4* gp = (const float4*)(warm   + rowoff + tbase);
    const int4*   mp = (const int4*)(mask     + rowoff + tbase);
#pragma unroll
    for (int q = 0; q < EPT / 4; ++q) {
      float4 xv = xp[q]; float4 gv = gp[q]; int4 mv = mp[q];
      xs[4*q+0] = xv.x; xs[4*q+1] = xv.y; xs[4*q+2] = xv.z; xs[4*q+3] = xv.w;
      gs[4*q+0] = gv.x; gs[4*q+1] = gv.y; gs[4*q+2] = gv.z; gs[4*q+3] = gv.w;
      ms[4*q+0] = mv.x ? 1.0f : 0.0f; ms[4*q+1] = mv.y ? 1.0f : 0.0f;
      ms[4*q+2] = mv.z ? 1.0f : 0.0f; ms[4*q+3] = mv.w ? 1.0f : 0.0f;
    }
  }

  __builtin_amdgcn_s_wait_tensorcnt(0);  // waves 1..7: TENSORcnt==0, no-op
  __syncthreads();                       // LDS tile visible to all waves

  // ---- mask-multiply pass: smh <- heatmap * mask ----
  for (int i = tid; i < SPAN; i += BDIM) {
    int p = s0 + i;
    if (p >= 0 && p < T_LEN)
      smh[i] *= (mask[rowoff + p] != 0) ? 1.0f : 0.0f;
  }
  __syncthreads();

  // ---- sliding-window max (W=41) + focal accumulation ----
  float a[EPT + 2 * W_R];   // 56 values: windows [j, j+40], j = 0..15
#pragma unroll
  for (int q = 0; q < (EPT + 2 * W_R) / 4; ++q) {
    float4 v = *(const float4*)&smh[tid * EPT + 4 * q];
    a[4*q+0] = v.x; a[4*q+1] = v.y; a[4*q+2] = v.z; a[4*q+3] = v.w;
  }
  // common part shared by all 16 windows: a[15..40]
  float common = a[15];
#pragma unroll
  for (int j = 16; j <= 40; ++j) common = fmaxf(common, a[j]);
  float L[15];              // L[j] = max(a[j..14])
  L[14] = a[14];
#pragma unroll
  for (int j = 13; j >= 0; --j) L[j] = fmaxf(a[j], L[j + 1]);
  float R[15];              // R[k] = max(a[41..41+k])
  R[0] = a[41];
#pragma unroll
  for (int k = 1; k < 15; ++k) R[k] = fmaxf(R[k - 1], a[41 + k]);

  float facc = 0.0f, npacc = 0.0f, msum = 0.0f, pksum = 0.0f;
#pragma unroll
  for (int j = 0; j < EPT; ++j) {
    float wmax = common;
    if (j < 15) wmax = fmaxf(wmax, L[j]);
    if (j > 0)  wmax = fmaxf(wmax, R[j - 1]);
    float mh = a[W_R + j];
    msum += mh;
    if (mh >= wmax) pksum += mh;   // ismax (window includes self)

    // CenterNet focal
    float x = xs[j], g = gs[j], mf = ms[j];
    float ax     = fabsf(x);
    float l1p    = log1pf(__expf(-ax));
    float logp   = fminf(x, 0.0f) - l1p;     // log sigmoid(x)
    float log1mp = fminf(-x, 0.0f) - l1p;    // log sigmoid(-x)
    float p      = 1.0f / (1.0f + __expf(-x));
    float pos    = (g >= 1.0f) ? 1.0f : 0.0f;
    float omp    = 1.0f - p;
    float g1v    = 1.0f - g; float g2v = g1v * g1v; float g4v = g2v * g2v;
    float pos_t  = omp * omp * logp * pos;
    float neg_t  = g4v * p * p * log1mp * (1.0f - pos);
    facc  += (pos_t + neg_t) * mf;
    npacc += pos * mf;
  }

  // ---- deterministic block tree reduction of 4 partials ----
  red[tid]            = facc;
  red[BDIM + tid]     = npacc;
  red[2 * BDIM + tid] = msum;
  red[3 * BDIM + tid] = pksum;
  __syncthreads();
  for (int s = BDIM >> 1; s > 0; s >>= 1) {
    if (tid < s) {
      red[tid]            += red[tid + s];
      red[BDIM + tid]     += red[BDIM + tid + s];
      red[2 * BDIM + tid] += red[2 * BDIM + tid + s];
      red[3 * BDIM + tid] += red[3 * BDIM + tid + s];
    }
    __syncthreads();
  }
  if (tid == 0) {
    float* pp = partials + (size_t)(b * NCHUNK + chunk) * 4;
    pp[0] = red[0]; pp[1] = red[BDIM]; pp[2] = red[2 * BDIM]; pp[3] = red[3 * BDIM];
  }
}

// =====================================================================
// Kernel 2: WMMA ones-matmul reduces partials[32][64][4] -> 128 row sums,
// then one wave32 per sample combines with the K=64 probe math.
// =====================================================================
__global__ __launch_bounds__(1024)
void finalize(const float* __restrict__ pcum,
              const float* __restrict__ refbp,
              const float* __restrict__ logS,
              const int*   __restrict__ centers,
              const int*   __restrict__ valid,
              const int*   __restrict__ nref,
              const float* __restrict__ partials,
              float* __restrict__ out)
{
  __shared__ float rowsum[B_N * 4];   // [b*4 + q]
  __shared__ float samp[B_N];
  const int  tid    = threadIdx.x;
  const int  w      = tid >> 5;       // wave id 0..31
  const int  lane   = tid & 31;
  const long rowoff = (long)w * T_LEN;

  // ---- Stage 1: D = A x ones + C chained over K=64 (16 x V_WMMA_F32_16X16X4_F32
  // per wave, waves 0..7). Row r = b*4 + q; M[r,c] = partials[(b*64+c)*4+q].
  // A layout (ISA): lanes 0-15 hold K={0,1}, lanes 16-31 hold K={2,3}. B = ones
  // is layout-invariant, so D[m,n] = sum_k A[m,k] for every n.
  if (w < 8) {
    const int row  = 16 * w + (lane & 15);
    const int rb_  = row >> 2;              // sample b
    const int rq_  = row & 3;               // quantity q
    const int koff = (lane >> 4) << 1;      // 0 or 2
    const float* base = partials + (size_t)rb_ * (NCHUNK * 4) + rq_;
    v8f d = {0.0f, 0.0f, 0.0f, 0.0f, 0.0f, 0.0f, 0.0f, 0.0f};
    v2f ones; ones[0] = 1.0f; ones[1] = 1.0f;
#pragma unroll
    for (int c4 = 0; c4 < 16; ++c4) {
      v2f av;
      av[0] = base[(4 * c4 + koff + 0) * 4];
      av[1] = base[(4 * c4 + koff + 1) * 4];
      d = __builtin_amdgcn_wmma_f32_16x16x4_f32(false, av, false, ones,
                                                (short)0, d, false, false);
    }
    // D VGPR r: lane 0 -> row 16w+r, lane 16 -> row 16w+8+r (column n=0).
    if (lane == 0 || lane == 16) {
      const int rbase = 16 * w + ((lane >> 4) << 3);
#pragma unroll
      for (int r = 0; r < 8; ++r) rowsum[rbase + r] = d[r];
    }
  }
  __syncthreads();

  const int b = w;                      // wave = sample
  const float fsum = rowsum[b * 4 + 0];
  const float np   = rowsum[b * 4 + 1];
  const float msum = rowsum[b * 4 + 2];
  const float pk   = rowsum[b * 4 + 3];

  const int*   ctr = centers + b * K_N;
  const float* rb  = refbp   + b * K_N;
  const float  r0  = rb[0];

  float vel = 0.0f, dpdr = 0.0f, dr2 = 0.0f, resol = 0.0f;
  float dpv[2], drv[2]; int has[2];
#pragma unroll
  for (int t = 0; t < 2; ++t) {
    int k = lane + 32 * t;
    int c = ctr[k]; c = c < 0 ? 0 : (c > T_LEN - 1 ? T_LEN - 1 : c);
    float pa = pcum[rowoff + c];
    float rk = rb[k];
    float z = (pa - rk) * (1.0f / 50.0f);
    vel += 0.5f * z * z;
    has[t] = (k < K_N - 1);
    dpv[t] = 0.0f; drv[t] = 0.0f;
    if (has[t]) {
      int cn = ctr[k + 1]; cn = cn < 0 ? 0 : (cn > T_LEN - 1 ? T_LEN - 1 : cn);
      float pan = pcum[rowoff + cn];
      float rkn = rb[k + 1];
      dpv[t] = pan - pa;
      drv[t] = rkn - rk;
      dpdr += dpv[t] * drv[t];
      dr2  += drv[t] * drv[t];
      float iv = fabsf(rkn - r0) - fabsf(rk - r0);   // ref_norm interval
      iv = fmaxf(iv, 1.0f);
      resol += fminf(iv * (1.0f / 511.0f), 1.0f);
    }
  }
  for (int m = 16; m >= 1; m >>= 1) {
    vel   += __shfl_xor(vel,   m, 32);
    dpdr  += __shfl_xor(dpdr,  m, 32);
    dr2   += __shfl_xor(dr2,   m, 32);
    resol += __shfl_xor(resol, m, 32);
  }

  float v = dpdr / (dr2 + 1e-6f);
  float S = fminf(fmaxf(__expf(logS[0]), 0.1f), 1000.0f);

  float bp = 0.0f;
#pragma unroll
  for (int t = 0; t < 2; ++t) {
    if (has[t]) {
      float var = S * fmaxf(drv[t], 1.0f);
      float r   = dpv[t] - v * drv[t];
      bp += 0.5f * logf(2.0f * PI_F * var) + r * r / (2.0f * var);
    }
  }
  for (int m = 16; m >= 1; m >>= 1) bp += __shfl_xor(bp, m, 32);

  if (lane == 0) {
    float focal_b = -fsum / fmaxf(np, 1.0f);
    float peaky   = 1.0f - pk / (msum + 1e-6f);
    float validf  = valid[b] ? 1.0f : 0.0f;
    float bp_b    = bp / (float)(K_N - 1);
    float vel_b   = vel / (float)K_N + logf(50.0f * sqrtf(2.0f * PI_F));
    float lv      = logf(fmaxf(v, 1e-6f));
    float stretch = 0.5f * lv * lv;
    int   nr      = nref[b]; if (nr < 2) nr = 2;
    float expected = fminf((float)nr, 1.0f + resol);
    float d        = msum - expected;
    float count_b  = d * d / (expected + 1.0f);
    // BLEND=0.5, LAM=0.8, all SCALE_*=1
    samp[b] = 0.5f * focal_b * validf + 0.5f * peaky
            + 0.8f * (bp_b + vel_b + count_b + stretch);
  }
  __syncthreads();
  if (tid == 0) {
    float s = 0.0f;
    for (int i = 0; i < B_N; ++i) s += samp[i];
    out[0] = s / (float)B_N;
  }
}

// =====================================================================
// launch
// =====================================================================
extern "C" void kernel_launch(void* const* d_in, const int* in_sizes, int n_in,
                              void* d_out, int out_size, void* d_ws, size_t ws_size,
                              hipStream_t stream) {
  const float* hm     = (const float*)d_in[0];   // pred_heatmap       [B,T] f32
  const float* pcum   = (const float*)d_in[1];   // pred_cumulative_bp [B,T] f32
  /* d_in[2] raw_velocity: unused by the reference */
  const float* logits = (const float*)d_in[3];   // pred_heatmap_logits [B,T] f32
  const float* warm   = (const float*)d_in[4];   // warmstart_heatmap   [B,T] f32
  const float* refbp  = (const float*)d_in[5];   // ref_bp [B,K] f32
  const float* logS   = (const float*)d_in[6];   // scalar f32
  const int*   mask   = (const int*)d_in[7];     // mask [B,T] (bool -> int32)
  const int*   ctr    = (const int*)d_in[8];     // gt_centers [B,K] (int)
  const int*   valid  = (const int*)d_in[9];     // warmstart_valid [B] (bool -> int32)
  const int*   nrefp  = (const int*)d_in[10];    // n_ref_probes [B] int32

  float* partials = (float*)d_ws;                // 32*64*4 floats = 32 KB

  dim3 grid1(NCHUNK, B_N);
  fused_row_pass<<<grid1, dim3(BDIM), 0, stream>>>(hm, mask, logits, warm, partials);
  finalize<<<dim3(1), dim3(1024), 0, stream>>>(pcum, refbp, logS, ctr, valid, nrefp,
                                               partials, (float*)d_out);
}